// GloRe_Unit_9337258902402
// MI455X (gfx1250) — compile-verified
//
#include <hip/hip_runtime.h>
#include <hip/hip_bf16.h>

typedef __attribute__((ext_vector_type(2))) float v2f;
typedef __attribute__((ext_vector_type(8))) float v8f;

#define L_TOT 32768
#define C_IN  128
#define NS    128
#define NN    64

// workspace layout (float offsets)
#define WS_G 0                         // 4 * 128*128
#define WS_R (4 * 128 * 128)           // 4 * 128
#define WS_A (WS_R + 4 * 128)          // 4 * 128*128
#define WS_D (WS_A + 4 * 128 * 128)    // 4 * 128

__global__ __launch_bounds__(256) void zero_ws_kernel(float* __restrict__ ws, int n) {
    int i = blockIdx.x * 256 + threadIdx.x;
    if (i < n) ws[i] = 0.0f;
}

// ---------------------------------------------------------------------------
// Kernel 1: per-batch Gram matrix G = X * X^T (128x128) and rowsum r = X * 1
// grid (64, 4), block 256 (8 waves). Each block handles 512 columns of L.
// ---------------------------------------------------------------------------
__global__ __launch_bounds__(256) void gram_kernel(const float* __restrict__ x,
                                                   float* __restrict__ ws) {
    const int b = blockIdx.y;
    const float* X = x + (size_t)b * C_IN * L_TOT;
    float* G = ws + WS_G + b * 128 * 128;
    float* R = ws + WS_R + b * 128;

    __shared__ float sX[128][36];   // 128 rows x 32 k-cols (+pad)

    const int tid  = threadIdx.x;
    const int wave = tid >> 5;
    const int lane = tid & 31;
    const int lr   = lane & 15;
    const int hs   = lane >> 4;          // half select
    const int col0 = blockIdx.x * 512;

    v8f acc[8];
    #pragma unroll
    for (int j = 0; j < 8; ++j)
        #pragma unroll
        for (int v = 0; v < 8; ++v) acc[j][v] = 0.0f;

    float rsum = 0.0f;
    const int rrow = tid & 127;
    const int rc0  = (tid >> 7) << 4;    // 0 or 16

    for (int stage = 0; stage < 16; ++stage) {
        __syncthreads();
        for (int i = tid; i < 128 * 32; i += 256) {
            int rr = i >> 5, cc = i & 31;
            sX[rr][cc] = X[(size_t)rr * L_TOT + col0 + (stage << 5) + cc];
        }
        __syncthreads();

        #pragma unroll
        for (int c = 0; c < 16; ++c) rsum += sX[rrow][rc0 + c];

        for (int k0 = 0; k0 < 32; k0 += 4) {
            const int kk = k0 + (hs << 1);
            // fragment for each 16-row band; identical gather serves as A and B
            v2f f[8];
            #pragma unroll
            for (int j = 0; j < 8; ++j) {
                f[j].x = sX[(j << 4) + lr][kk];
                f[j].y = sX[(j << 4) + lr][kk + 1];
            }
            v2f a;
            a.x = sX[(wave << 4) + lr][kk];
            a.y = sX[(wave << 4) + lr][kk + 1];
            #pragma unroll
            for (int j = 0; j < 8; ++j) {
                acc[j] = __builtin_amdgcn_wmma_f32_16x16x4_f32(
                    false, a, false, f[j], (short)0, acc[j], false, false);
            }
        }
    }

    // accumulate partial Gram into global (C/D layout: lane<16 -> M=v, else M=v+8)
    #pragma unroll
    for (int j = 0; j < 8; ++j)
        #pragma unroll
        for (int v = 0; v < 8; ++v) {
            int rowt = (wave << 4) + (hs ? (v + 8) : v);
            atomicAdd(&G[rowt * 128 + (j << 4) + lr], acc[j][v]);
        }
    atomicAdd(&R[rrow], rsum);
}

// ---------------------------------------------------------------------------
// Kernel 2: the graph middle. One block per batch (grid 4, block 512).
// Consumes G, r; produces per-batch A (128x128, BN-folded) and d (128).
// ---------------------------------------------------------------------------
__global__ __launch_bounds__(512) void middle_kernel(
    float* __restrict__ ws,
    const float* __restrict__ Ws, const float* __restrict__ bs,
    const float* __restrict__ Wp, const float* __restrict__ bp,
    const float* __restrict__ W1, const float* __restrict__ b1,
    const float* __restrict__ W2,
    const float* __restrict__ lin1,
    const float* __restrict__ a1l, const float* __restrict__ a1r,
    const float* __restrict__ lin2,
    const float* __restrict__ a2l, const float* __restrict__ a2r,
    const float* __restrict__ We,
    const float* __restrict__ gam, const float* __restrict__ bet,
    const float* __restrict__ rmean, const float* __restrict__ rvar)
{
    const int b   = blockIdx.x;
    const int tid = threadIdx.x;
    const float* G = ws + WS_G + b * 128 * 128;
    const float* R = ws + WS_R + b * 128;
    float* Aout = ws + WS_A + b * 128 * 128;
    float* Dout = ws + WS_D + b * 128;

    __shared__ float bufA[128 * 64];
    __shared__ float bufB[128 * 64];
    __shared__ float sW1[64 * 64];
    __shared__ float sal[64 * 64];
    __shared__ float sg[64 * 16];
    __shared__ float sg1[64 * 16];
    __shared__ float sel[64 * 8], ser[64 * 8];
    __shared__ float sel2[64], ser2[64];
    __shared__ float su[128], sv[64];
    __shared__ float sr[128];

    for (int i = tid; i < 64 * 64; i += 512) sW1[i] = W1[i];
    if (tid < 128) sr[tid] = R[tid];
    __syncthreads();

    // u = Ws @ r, v = Wp @ r
    if (tid < 128) {
        float a = 0.f;
        for (int c = 0; c < 128; ++c) a += Ws[tid * 128 + c] * sr[c];
        su[tid] = a;
    } else if (tid < 192) {
        int n = tid - 128;
        float a = 0.f;
        for (int c = 0; c < 128; ++c) a += Wp[n * 128 + c] * sr[c];
        sv[n] = a;
    }
    // T1 = G @ Wp^T  -> bufA[c][n]
    for (int idx = tid; idx < 128 * 64; idx += 512) {
        int c = idx >> 6, n = idx & 63;
        float a = 0.f;
        for (int k = 0; k < 128; ++k) a += G[c * 128 + k] * Wp[n * 128 + k];
        bufA[idx] = a;
    }
    __syncthreads();
    // xns = Ws @ T1 + rank-1 terms -> bufB[s][n]
    for (int idx = tid; idx < 128 * 64; idx += 512) {
        int s = idx >> 6, n = idx & 63;
        float a = 0.f;
        for (int k = 0; k < 128; ++k) a += Ws[s * 128 + k] * bufA[k * 64 + n];
        a += su[s] * bp[n] + bs[s] * sv[n] + 32768.0f * bs[s] * bp[n];
        bufB[idx] = a;
    }
    __syncthreads();
    // hh[s][n] = sum_j W1[n][j] xns[s][j] + b1[n] + xns[s][n] -> bufA
    for (int idx = tid; idx < 128 * 64; idx += 512) {
        int s = idx >> 6, n = idx & 63;
        float a = 0.f;
        for (int j = 0; j < 64; ++j) a += sW1[n * 64 + j] * bufB[s * 64 + j];
        bufA[idx] = a + b1[n] + bufB[idx];
    }
    __syncthreads();
    // h3 = W2 @ relu(hh) -> bufB[s][n]
    for (int idx = tid; idx < 128 * 64; idx += 512) {
        int s = idx >> 6, n = idx & 63;
        float a = 0.f;
        for (int t = 0; t < 128; ++t) {
            float hv = bufA[t * 64 + n];
            a += W2[s * 128 + t] * (hv > 0.f ? hv : 0.f);
        }
        bufB[idx] = a;
    }
    __syncthreads();
    // GAT1 linear: g[n][o] = sum_f lin1[o][f] * h3[f][n]
    for (int idx = tid; idx < 64 * 16; idx += 512) {
        int n = idx >> 4, o = idx & 15;
        float a = 0.f;
        for (int f = 0; f < 128; ++f) a += lin1[o * 128 + f] * bufB[f * 64 + n];
        sg[idx] = a;
    }
    __syncthreads();
    // attention scalars
    {
        int n = tid >> 3, h = tid & 7;  // tid < 512 always
        sel[tid] = sg[n * 16 + 2 * h] * a1l[2 * h] + sg[n * 16 + 2 * h + 1] * a1l[2 * h + 1];
        ser[tid] = sg[n * 16 + 2 * h] * a1r[2 * h] + sg[n * 16 + 2 * h + 1] * a1r[2 * h + 1];
    }
    __syncthreads();
    // GAT1 softmax + aggregate + ELU -> sg1
    {
        int i = tid >> 3, h = tid & 7;
        float eli = sel[i * 8 + h];
        float mx = -1e30f;
        for (int j = 0; j < 64; ++j) {
            float e;
            if (sW1[i * 64 + j] != 0.f) { e = eli + ser[j * 8 + h]; e = (e >= 0.f) ? e : 0.2f * e; }
            else e = -1e9f;
            mx = fmaxf(mx, e);
        }
        float den = 0.f, o0 = 0.f, o1 = 0.f;
        for (int j = 0; j < 64; ++j) {
            float e;
            if (sW1[i * 64 + j] != 0.f) { e = eli + ser[j * 8 + h]; e = (e >= 0.f) ? e : 0.2f * e; }
            else e = -1e9f;
            float w = __expf(e - mx);
            den += w;
            o0 += w * sg[j * 16 + 2 * h];
            o1 += w * sg[j * 16 + 2 * h + 1];
        }
        float r0 = o0 / den, r1 = o1 / den;
        sg1[i * 16 + 2 * h]     = (r0 > 0.f) ? r0 : __expf(r0) - 1.f;
        sg1[i * 16 + 2 * h + 1] = (r1 > 0.f) ? r1 : __expf(r1) - 1.f;
    }
    __syncthreads();
    // GAT2 linear: g2lin[n][d] = sum_f sg1[n][f] * lin2[d][f] -> bufA[n*128+d]
    for (int idx = tid; idx < 64 * 128; idx += 512) {
        int n = idx >> 7, d = idx & 127;
        float a = 0.f;
        #pragma unroll
        for (int f = 0; f < 16; ++f) a += sg1[n * 16 + f] * lin2[d * 16 + f];
        bufA[idx] = a;
    }
    __syncthreads();
    if (tid < 64) {
        float a = 0.f;
        for (int d = 0; d < 128; ++d) a += bufA[tid * 128 + d] * a2l[d];
        sel2[tid] = a;
    } else if (tid < 128) {
        int n = tid - 64;
        float a = 0.f;
        for (int d = 0; d < 128; ++d) a += bufA[n * 128 + d] * a2r[d];
        ser2[n] = a;
    }
    __syncthreads();
    // GAT2 attention rows -> sal (normalized)
    if (tid < 64) {
        int i = tid;
        float mx = -1e30f;
        for (int j = 0; j < 64; ++j) {
            float e;
            if (sW1[i * 64 + j] != 0.f) { e = sel2[i] + ser2[j]; e = (e >= 0.f) ? e : 0.2f * e; }
            else e = -1e9f;
            mx = fmaxf(mx, e);
        }
        float den = 0.f;
        for (int j = 0; j < 64; ++j) {
            float e;
            if (sW1[i * 64 + j] != 0.f) { e = sel2[i] + ser2[j]; e = (e >= 0.f) ? e : 0.2f * e; }
            else e = -1e9f;
            float w = __expf(e - mx);
            sal[i * 64 + j] = w;
            den += w;
        }
        float inv = 1.f / den;
        for (int j = 0; j < 64; ++j) sal[i * 64 + j] *= inv;
    }
    __syncthreads();
    // g2[i][d] = sum_j sal[i][j] * g2lin[j][d] -> bufB[i*128+d]
    for (int idx = tid; idx < 64 * 128; idx += 512) {
        int i = idx >> 7, d = idx & 127;
        float a = 0.f;
        for (int j = 0; j < 64; ++j) a += sal[i * 64 + j] * bufA[j * 128 + d];
        bufB[idx] = a;
    }
    __syncthreads();
    // M = We @ x_n_rel -> bufA[c*64+n]   (x_n_rel[s][n] = g2[n][s])
    for (int idx = tid; idx < 128 * 64; idx += 512) {
        int c = idx >> 6, n = idx & 63;
        float a = 0.f;
        for (int s = 0; s < 128; ++s) a += We[c * 128 + s] * bufB[n * 128 + s];
        bufA[idx] = a;
    }
    __syncthreads();
    // A = diag(scale) * (M @ Wp),  d = scale*(M@bp - rmean) + beta
    for (int idx = tid; idx < 128 * 128; idx += 512) {
        int c = idx >> 7, cc = idx & 127;
        float a = 0.f;
        for (int n = 0; n < 64; ++n) a += bufA[c * 64 + n] * Wp[n * 128 + cc];
        float sc = gam[c] * rsqrtf(rvar[c] + 1e-4f);
        Aout[idx] = sc * a;
    }
    if (tid < 128) {
        int c = tid;
        float q = 0.f;
        for (int n = 0; n < 64; ++n) q += bufA[c * 64 + n] * bp[n];
        float sc = gam[c] * rsqrtf(rvar[c] + 1e-4f);
        Dout[c] = sc * (q - rmean[c]) + bet[c];
    }
}

// ---------------------------------------------------------------------------
// Kernel 3: out[:,l] = x[:,l] + A @ x[:,l] + d.  GEMM 128x128 @ 128xL + eltwise.
// grid (256, 4), block 256 (8 waves, each owns one 16-col tile across 8 M-tiles)
// ---------------------------------------------------------------------------
__global__ __launch_bounds__(256) void apply_kernel(const float* __restrict__ x,
                                                    const float* __restrict__ ws,
                                                    float* __restrict__ out) {
    const int b = blockIdx.y;
    const float* X = x + (size_t)b * C_IN * L_TOT;
    float* O = out + (size_t)b * C_IN * L_TOT;
    const float* Am = ws + WS_A + b * 128 * 128;
    const float* Dv = ws + WS_D + b * 128;

    __shared__ float sA[128 * 129];
    __shared__ float sXt[128 * 136];
    __shared__ float sd[128];

    const int tid  = threadIdx.x;
    const int wave = tid >> 5;
    const int lane = tid & 31;
    const int lr   = lane & 15;
    const int hs   = lane >> 4;
    const int col0 = blockIdx.x * 128;

    for (int i = tid; i < 128 * 128; i += 256) {
        int r = i >> 7, c = i & 127;
        sA[r * 129 + c]  = Am[i];
        sXt[r * 136 + c] = X[(size_t)r * L_TOT + col0 + c];
    }
    if (tid < 128) sd[tid] = Dv[tid];
    __syncthreads();

    v8f acc[8];
    #pragma unroll
    for (int m = 0; m < 8; ++m)
        #pragma unroll
        for (int v = 0; v < 8; ++v) acc[m][v] = 0.0f;

    const int cb = wave << 4;
    for (int k0 = 0; k0 < 128; k0 += 4) {
        const int kk = k0 + (hs << 1);
        v2f bf;
        bf.x = sXt[kk * 136 + cb + lr];
        bf.y = sXt[(kk + 1) * 136 + cb + lr];
        #pragma unroll
        for (int mt = 0; mt < 8; ++mt) {
            const int row = (mt << 4) + lr;
            v2f af;
            af.x = sA[row * 129 + kk];
            af.y = sA[row * 129 + kk + 1];
            acc[mt] = __builtin_amdgcn_wmma_f32_16x16x4_f32(
                false, af, false, bf, (short)0, acc[mt], false, false);
        }
    }

    #pragma unroll
    for (int mt = 0; mt < 8; ++mt)
        #pragma unroll
        for (int v = 0; v < 8; ++v) {
            int rowt = (mt << 4) + (hs ? (v + 8) : v);
            int col  = cb + lr;
            O[(size_t)rowt * L_TOT + col0 + col] =
                acc[mt][v] + sXt[rowt * 136 + col] + sd[rowt];
        }
}

// ---------------------------------------------------------------------------
extern "C" void kernel_launch(void* const* d_in, const int* in_sizes, int n_in,
                              void* d_out, int out_size, void* d_ws, size_t ws_size,
                              hipStream_t stream) {
    (void)in_sizes; (void)n_in; (void)out_size; (void)ws_size;
    const float* x    = (const float*)d_in[0];
    const float* Ws   = (const float*)d_in[1];
    const float* bs   = (const float*)d_in[2];
    const float* Wp   = (const float*)d_in[3];
    const float* bp   = (const float*)d_in[4];
    const float* W1   = (const float*)d_in[5];
    const float* b1   = (const float*)d_in[6];
    const float* W2   = (const float*)d_in[7];
    const float* lin1 = (const float*)d_in[8];
    const float* a1l  = (const float*)d_in[9];
    const float* a1r  = (const float*)d_in[10];
    const float* lin2 = (const float*)d_in[11];
    const float* a2l  = (const float*)d_in[12];
    const float* a2r  = (const float*)d_in[13];
    const float* We   = (const float*)d_in[14];
    const float* gam  = (const float*)d_in[15];
    const float* bet  = (const float*)d_in[16];
    const float* rmean= (const float*)d_in[17];
    const float* rvar = (const float*)d_in[18];
    float* out = (float*)d_out;
    float* ws  = (float*)d_ws;

    const int nzero = WS_A;  // G + r regions must start at 0 (accumulated via atomics)
    zero_ws_kernel<<<(nzero + 255) / 256, 256, 0, stream>>>(ws, nzero);
    gram_kernel<<<dim3(64, 4), 256, 0, stream>>>(x, ws);
    middle_kernel<<<dim3(4), 512, 0, stream>>>(ws, Ws, bs, Wp, bp, W1, b1, W2,
                                               lin1, a1l, a1r, lin2, a2l, a2r,
                                               We, gam, bet, rmean, rvar);
    apply_kernel<<<dim3(256, 4), 256, 0, stream>>>(x, ws, out);
}